// ExodusNet_81587198754936
// MI455X (gfx1250) — compile-verified
//
#include <hip/hip_runtime.h>
#include <hip/hip_bf16.h>

typedef __attribute__((ext_vector_type(2))) float v2f;
typedef __attribute__((ext_vector_type(8))) float v8f;

#define NN 16384
#define TT_STEPS 100
// alpha = exp(-1/10) computed in double, rounded to f32 (matches jnp promotion)
#define ALPHA_F 0.9048374180359595f
#define ONE_M_ALPHA_F 0.09516258196404048f

// ---------------------------------------------------------------------------
// Kernel 1: weighted[n,t] = sum_{k=0..31} x[n,k,t] * W[k]  via f32 WMMA.
// One wave32 per neuron. x flat index: n*3200 + k*100 + t.
// A-matrix (16x4 f32): lane holds row M=lane%16; V0/V1 = K pair selected by
// lane half (lanes 0-15 -> K{0,1}, lanes 16-31 -> K{2,3}).
// B-matrix (4x16 f32): W broadcast into every column; lane half selects the
// same K pair, so each lane supplies (W[k], W[k+1]).
// D (16x16 f32): VGPR r holds M=r (lanes 0-15) / M=r+8 (lanes 16-31); all
// columns identical, so lanes 0-7 / 16-23 extract the 16 per-row results.
// Results are written into `out` (same size as final output) as scratch.
// ---------------------------------------------------------------------------
__global__ __launch_bounds__(256) void exodus_dense_wmma(
    const float* __restrict__ x, const float* __restrict__ W,
    float* __restrict__ out) {
  const int lane = threadIdx.x & 31;
  const int wave = threadIdx.x >> 5;
  const int n    = blockIdx.x * 8 + wave;
  const int half = lane >> 4;   // 0: lanes 0-15, 1: lanes 16-31
  const int lm   = lane & 15;   // A-matrix row within tile

  // Preload B (column-broadcast of W) for all 8 K-tiles: 16 VGPRs.
  v2f bv[8];
#pragma unroll
  for (int i = 0; i < 8; ++i) {
    const int k = 4 * i + 2 * half;
    bv[i].x = W[k];
    bv[i].y = W[k + 1];
  }

  const long nbase = (long)n * 3200;
  float* orow = out + (long)n * TT_STEPS;

#pragma unroll 1
  for (int tt = 0; tt < 7; ++tt) {           // 7 tiles of 16 timesteps (112, clamped)
    const int trow_ld = tt * 16 + lm;
    const int tcl     = trow_ld < (TT_STEPS - 1) ? trow_ld : (TT_STEPS - 1);
    const float* pa   = x + nbase + tcl + 200 * half; // + k-pair base offset

    v8f c = {0.f, 0.f, 0.f, 0.f, 0.f, 0.f, 0.f, 0.f};
#pragma unroll
    for (int i = 0; i < 8; ++i) {            // K = 32 in 8 steps of 4
      v2f a;
      a.x = pa[i * 400];                     // K = 4i + 2*half
      a.y = pa[i * 400 + 100];               // K = 4i + 2*half + 1
      c = __builtin_amdgcn_wmma_f32_16x16x4_f32(
          /*neg_a=*/false, a, /*neg_b=*/false, bv[i],
          /*c_mod=*/(short)0, c, /*reuse_a=*/false, /*reuse_b=*/false);
    }

    // Extract per-row results (all 16 columns of D are equal).
    const int r = lane & 7;
    float val = c[0];
    val = (r == 1) ? c[1] : val;
    val = (r == 2) ? c[2] : val;
    val = (r == 3) ? c[3] : val;
    val = (r == 4) ? c[4] : val;
    val = (r == 5) ? c[5] : val;
    val = (r == 6) ? c[6] : val;
    val = (r == 7) ? c[7] : val;

    const int trow = tt * 16 + (half ? 8 : 0) + r;
    if ((lm < 8) && (trow < TT_STEPS)) orow[trow] = val;
  }
}

// ---------------------------------------------------------------------------
// Kernel 2: in-place LIF scan per neuron over T=100 steps.
//   v = alpha*v + (1-alpha)*i ; s = (v >= 1) ; v -= s (membrane subtract)
// Reads the weighted inputs kernel 1 left in `out` (hot in L2), overwrites
// with the spike train. Deterministic: kernel 1 fully rewrites `out` first.
// ---------------------------------------------------------------------------
__global__ __launch_bounds__(256) void exodus_lif_scan(float* __restrict__ out) {
  const int n = blockIdx.x * 256 + threadIdx.x;
  float* p = out + (long)n * TT_STEPS;
  float v = 0.0f;
#pragma unroll 4
  for (int t = 0; t < TT_STEPS; ++t) {
    const float inp = p[t];
    v = ALPHA_F * v + ONE_M_ALPHA_F * inp;
    const float s = (v >= 1.0f) ? 1.0f : 0.0f;
    p[t] = s;
    v -= s;
  }
}

extern "C" void kernel_launch(void* const* d_in, const int* in_sizes, int n_in,
                              void* d_out, int out_size, void* d_ws, size_t ws_size,
                              hipStream_t stream) {
  const float* x = (const float*)d_in[0];   // [N, 2, 4, 4, T] f32
  const float* W = (const float*)d_in[1];   // [1, 2, 4, 4] f32 (32 elems)
  float* out = (float*)d_out;               // [N, 1, 1, 1, T] f32

  // 16384 neurons, one wave32 each: 2048 blocks x 8 waves.
  exodus_dense_wmma<<<dim3(NN / 8), dim3(256), 0, stream>>>(x, W, out);
  // 16384 neurons, one thread each.
  exodus_lif_scan<<<dim3(NN / 256), dim3(256), 0, stream>>>(out);
}